// LE_Conv_52888227283032
// MI455X (gfx1250) — compile-verified
//
#include <hip/hip_runtime.h>

typedef __attribute__((ext_vector_type(2))) float v2f;
typedef __attribute__((ext_vector_type(4))) float v4f;
typedef __attribute__((ext_vector_type(8))) float v8f;

#define D 128          // D_IN == D_OUT
#define CDIM 32        // community size
#define TILE_ROWS 128  // rows of N per workgroup (8 waves x 16)
#define WG_THREADS 256
#define LDS_STRIDE 260 // 256 + 4 pad: stride%64==4 -> conflict-free b64 A-frag loads

// Build combined weight Bext = [ (W1 + 32*W2)^T ; (-W3)^T ]  (K=256 x O=128),
// stored pre-swizzled into the exact WMMA B-fragment layout:
//   bswz[ct][k][lane][j] = Bext[4k + 2*(lane/16) + j][16*ct + lane%16]
__global__ void build_bswz_kernel(const float* __restrict__ W1,
                                  const float* __restrict__ W2,
                                  const float* __restrict__ W3,
                                  float* __restrict__ bswz) {
    int tid = blockIdx.x * blockDim.x + threadIdx.x;
    if (tid >= 8 * 64 * 32 * 2) return;
    int j    = tid & 1;
    int lane = (tid >> 1) & 31;
    int k    = (tid >> 6) & 63;
    int ct   = tid >> 12;
    int o  = ct * 16 + (lane & 15);
    int kk = 4 * k + 2 * (lane >> 4) + j;
    float v;
    if (kk < D) v = W1[o * D + kk] + 32.0f * W2[o * D + kk];
    else        v = -W3[o * D + (kk - D)];
    bswz[tid] = v;  // index == ct*4096 + k*64 + lane*2 + j
}

__launch_bounds__(WG_THREADS)
__global__ void fused_kernel(const float* __restrict__ src,
                             const float* __restrict__ comm,
                             const float* __restrict__ bswz,
                             float* __restrict__ out,
                             int N) {
    __shared__ float X[TILE_ROWS * LDS_STRIDE];  // [row][K=256 (+pad)] extended operand

    const int tid   = threadIdx.x;
    const int tile0 = blockIdx.x * TILE_ROWS;

    // ---- Phase 1a: stage src rows -> X[row][0:128) ----
    for (int s = tid; s < TILE_ROWS * (D / 4); s += WG_THREADS) {
        int row = s >> 5;
        int d4  = s & 31;
        int n   = tile0 + row;
        v4f v = {0.0f, 0.0f, 0.0f, 0.0f};
        if (n < N) v = *(const v4f*)(src + (size_t)n * D + d4 * 4);
        *(v4f*)(&X[row * LDS_STRIDE + d4 * 4]) = v;
    }

    // ---- Phase 1b: stream-reduce comm over C -> X[row][128:256) ----
    // 819 MB total across the grid: this is THE bandwidth-critical loop.
    for (int s = tid; s < TILE_ROWS * (D / 4); s += WG_THREADS) {
        int row = s >> 5;
        int d4  = s & 31;
        int n   = tile0 + row;
        v4f acc = {0.0f, 0.0f, 0.0f, 0.0f};
        if (n < N) {
            const v4f* p = (const v4f*)(comm + (size_t)n * (CDIM * D) + d4 * 4);
            #pragma unroll 8
            for (int c = 0; c < CDIM; ++c) {
                v4f v = __builtin_nontemporal_load(p + c * (D / 4));
                acc += v;
            }
        }
        *(v4f*)(&X[row * LDS_STRIDE + D + d4 * 4]) = acc;
    }
    __syncthreads();

    // ---- Phase 2: per-wave 16x128 output tile via V_WMMA_F32_16X16X4_F32 ----
    const int wave = tid >> 5;
    const int lane = tid & 31;
    const int row0 = tile0 + wave * 16;
    if (row0 >= N) return;            // wave-uniform: EXEC stays all-1s for WMMA

    const int m    = lane & 15;       // A-frag row / store column index
    const int half = lane >> 4;       // K sub-offset select (0 or 1)
    const bool full = (row0 + 15) < N;  // wave-uniform; always true when N % 16 == 0
    const float* arow = &X[(wave * 16 + m) * LDS_STRIDE + 2 * half];

    for (int ct = 0; ct < 8; ++ct) {  // 8 column tiles of 16
        v8f acc = {0.f,0.f,0.f,0.f,0.f,0.f,0.f,0.f};
        const float* bbase = bswz + ct * 4096 + lane * 2;
        #pragma unroll 4
        for (int k = 0; k < 64; ++k) {                 // K = 256, 4 per WMMA
            v2f a = *(const v2f*)(arow + 4 * k);       // ds_load_b64, bank-conflict-free
            v2f b = *(const v2f*)(bbase + k * 64);     // global_load_b64, L2-hot
            acc = __builtin_amdgcn_wmma_f32_16x16x4_f32(
                false, a, false, b, (short)0, acc, false, false);
        }
        int o = ct * 16 + m;
        float* obase = out + (size_t)(row0 + 8 * half) * D + o;
        if (full) {
            // Hot path: branch-free stores (D layout: M = j + 8*half, N = m).
            #pragma unroll
            for (int j = 0; j < 8; ++j)
                __builtin_nontemporal_store(acc[j], obase + (size_t)j * D);
        } else {
            #pragma unroll
            for (int j = 0; j < 8; ++j) {
                int n = row0 + j + 8 * half;
                if (n < N)
                    __builtin_nontemporal_store(acc[j], out + (size_t)n * D + o);
            }
        }
    }
}

extern "C" void kernel_launch(void* const* d_in, const int* in_sizes, int n_in,
                              void* d_out, int out_size, void* d_ws, size_t ws_size,
                              hipStream_t stream) {
    const float* src  = (const float*)d_in[0];
    const float* comm = (const float*)d_in[1];
    const float* W1   = (const float*)d_in[2];
    const float* W2   = (const float*)d_in[3];
    const float* W3   = (const float*)d_in[4];
    float* out  = (float*)d_out;
    float* bswz = (float*)d_ws;     // 32768 floats = 128 KB of scratch

    const int N = in_sizes[0] / D;  // 50000

    build_bswz_kernel<<<(8 * 64 * 32 * 2 + 255) / 256, 256, 0, stream>>>(W1, W2, W3, bswz);

    const int grid = (N + TILE_ROWS - 1) / TILE_ROWS;  // 391
    fused_kernel<<<grid, WG_THREADS, 0, stream>>>(src, comm, bswz, out, N);
}